// SetAttention_25039659335843
// MI455X (gfx1250) — compile-verified
//
#include <hip/hip_runtime.h>
#include <math.h>

typedef __attribute__((ext_vector_type(2))) float v2f;
typedef __attribute__((ext_vector_type(8))) float v8f;

#define LNEPS 1e-5f
#define EPSC 1e-8f
#define SCALEF 0.0625f   // 256^-0.5

static __device__ __forceinline__ v8f wmma_f32(v2f a, v2f b, v8f c) {
  // D = A(16x4 f32) * B(4x16 f32) + C(16x16 f32)
  return __builtin_amdgcn_wmma_f32_16x16x4_f32(false, a, false, b, (short)0, c,
                                               false, false);
}

// Async global->LDS copy of 16 bytes per lane (CDNA5 GLOBAL_LOAD_ASYNC_TO_LDS,
// tracked by ASYNCcnt). ldsaddr = wave-relative LDS byte address (= low 32
// bits of the generic shared pointer, ISA 10.2 aperture mapping).
static __device__ __forceinline__ void async_copy_b128(void* lds_dst,
                                                       const void* gsrc) {
  unsigned la = (unsigned)(size_t)lds_dst;
  asm volatile("global_load_async_to_lds_b128 %0, %1, off" ::"v"(la), "v"(gsrc)
               : "memory");
}

static __device__ __forceinline__ void wait_asynccnt0() {
  asm volatile("s_wait_asynccnt 0x0" ::: "memory");
}

__device__ __forceinline__ float block_sum(float v, float* red) {
  int t = threadIdx.x;
  red[t] = v;
  __syncthreads();
  for (int s = 128; s > 0; s >>= 1) {
    if (t < s) red[t] += red[t + s];
    __syncthreads();
  }
  float r = red[0];
  __syncthreads();
  return r;
}

// ---------------------------------------------------------------------------
// Kernel 1: LayerNorm(inputs) then k = xn@Wk^T+bk, v = xn@Wv^T+bv (f32 WMMA)
// 32 rows per block staged into LDS with async b128 copies; 8 waves; wave
// handles 4 column-tiles (16 cols) of the 512-wide (k|v) output with 2
// row-tile accumulators each.
// ---------------------------------------------------------------------------
extern "C" __global__ void __launch_bounds__(256)
ln_kv_kernel(const float* __restrict__ in, const float* __restrict__ g,
             const float* __restrict__ be, const float* __restrict__ Wk,
             const float* __restrict__ bk, const float* __restrict__ Wv,
             const float* __restrict__ bv, float* __restrict__ kout,
             float* __restrict__ vout) {
  __shared__ float xlds[32 * 256];
  __shared__ float red[256];
  __shared__ float mu_s[32];
  __shared__ float rs_s[32];
  const int t = threadIdx.x;
  const long row0 = (long)blockIdx.x * 32;

  // Async stage 32 rows (32KB): 2048 float4 segments, 8 per thread.
  {
    const float* gsrc = in + row0 * 256;
    for (int i = 0; i < 8; ++i) {
      int f4 = i * 256 + t;
      async_copy_b128(xlds + f4 * 4, gsrc + f4 * 4);
    }
    wait_asynccnt0();
  }
  __syncthreads();

  // LayerNorm: 8 threads per row
  const int lrow = t >> 3, part = t & 7;
  {
    float s = 0.f;
    for (int kk = 0; kk < 32; ++kk) s += xlds[lrow * 256 + part * 32 + kk];
    red[t] = s;
  }
  __syncthreads();
  if (part == 0) {
    float s = 0.f;
    for (int p = 0; p < 8; ++p) s += red[(lrow << 3) + p];
    mu_s[lrow] = s * (1.f / 256.f);
  }
  __syncthreads();
  {
    float mu = mu_s[lrow];
    float s = 0.f;
    for (int kk = 0; kk < 32; ++kk) {
      float d = xlds[lrow * 256 + part * 32 + kk] - mu;
      s += d * d;
    }
    red[t] = s;
  }
  __syncthreads();
  if (part == 0) {
    float s = 0.f;
    for (int p = 0; p < 8; ++p) s += red[(lrow << 3) + p];
    rs_s[lrow] = rsqrtf(s * (1.f / 256.f) + LNEPS);
  }
  __syncthreads();
  {
    float mu = mu_s[lrow], rs = rs_s[lrow];
    for (int kk = 0; kk < 32; ++kk) {
      int d = part * 32 + kk;
      int idx = lrow * 256 + d;
      xlds[idx] = (xlds[idx] - mu) * rs * g[d] + be[d];
    }
  }
  __syncthreads();

  // Dual GEMM via v_wmma_f32_16x16x4_f32
  const int lane = t & 31, w = t >> 5;
  const int mrow = lane & 15;            // M (for A) / N (for B,C)
  const int klo = (lane >> 4) << 1;      // K sub-offset per lane half
  for (int c = 0; c < 4; ++c) {
    int colbase = (w * 4 + c) * 16;      // 0..511
    const float* Wsel = (colbase < 256) ? Wk : Wv;
    const float* bsel = (colbase < 256) ? bk : bv;
    float* outsel = (colbase < 256) ? kout : vout;
    int cin = (colbase & 255) + mrow;
    float bias = bsel[cin];
    v8f acc0 = {};
    v8f acc1 = {};
    const float* wp = Wsel + (long)cin * 256 + klo;
    const float* x0 = xlds + mrow * 256 + klo;
    const float* x1 = x0 + 16 * 256;
    for (int kb = 0; kb < 256; kb += 4) {
      v2f bf = *(const v2f*)(wp + kb);
      v2f f0 = *(const v2f*)(x0 + kb);
      v2f f1 = *(const v2f*)(x1 + kb);
      acc0 = wmma_f32(f0, bf, acc0);
      acc1 = wmma_f32(f1, bf, acc1);
    }
    long orow = row0 + ((lane >> 4) << 3);  // C layout: M = r + 8*(lane/16)
    for (int r = 0; r < 8; ++r) {
      outsel[(orow + r) * 256 + cin] = acc0[r] + bias;
      outsel[(orow + 16 + r) * 256 + cin] = acc1[r] + bias;
    }
  }
}

// ---------------------------------------------------------------------------
// Kernel 2: sets = mu + sigma * noise
// ---------------------------------------------------------------------------
extern "C" __global__ void __launch_bounds__(256)
init_sets_kernel(const float* __restrict__ noise, const float* __restrict__ mu,
                 const float* __restrict__ sg, float* __restrict__ sets) {
  int idx = blockIdx.x * 256 + threadIdx.x;
  int d = idx & 255;
  sets[idx] = mu[d] + sg[d] * noise[idx];
}

// ---------------------------------------------------------------------------
// Kernel 3: q = LN(sets)@Wq^T + bq  (128 rows; small, VALU)
// ---------------------------------------------------------------------------
extern "C" __global__ void __launch_bounds__(256)
qproj_kernel(const float* __restrict__ sets, const float* __restrict__ g,
             const float* __restrict__ be, const float* __restrict__ Wq,
             const float* __restrict__ bq, float* __restrict__ qout) {
  __shared__ float xn[256];
  __shared__ float red[256];
  int row = blockIdx.x, t = threadIdx.x;
  float v = sets[(long)row * 256 + t];
  float mu = block_sum(v, red) * (1.f / 256.f);
  float d = v - mu;
  float var = block_sum(d * d, red) * (1.f / 256.f);
  xn[t] = d * rsqrtf(var + LNEPS) * g[t] + be[t];
  __syncthreads();
  float acc = bq[t];
  const float* wr = Wq + (long)t * 256;
  for (int k = 0; k < 256; ++k) acc += xn[k] * wr[k];
  qout[(long)row * 256 + t] = acc;
}

// ---------------------------------------------------------------------------
// Kernel 4: zero U and S accumulators
// ---------------------------------------------------------------------------
extern "C" __global__ void __launch_bounds__(256)
zero_us_kernel(float* __restrict__ U, float* __restrict__ S) {
  int idx = blockIdx.x * 256 + threadIdx.x;
  U[idx] = 0.f;
  if (idx < 128) S[idx] = 0.f;
}

// ---------------------------------------------------------------------------
// Kernel 5: per (batch, 512-col chunk of j):
//   dots = q@k^T (WMMA, M padded 8->16), softmax over slot axis (pure in-lane
//   op thanks to the C-matrix layout: lanes 0-15 hold all 8 slots in regs),
//   unnormalized attn (+eps) -> LDS, then U_partial = attn@v (WMMA) and
//   S_partial = rowsum(attn), atomically accumulated into global U/S.
// ---------------------------------------------------------------------------
extern "C" __global__ void __launch_bounds__(256)
attn_kernel(const float* __restrict__ q, const float* __restrict__ kbuf,
            const float* __restrict__ vbuf, float* __restrict__ Ubuf,
            float* __restrict__ Sbuf) {
  __shared__ float qpad[16 * 256];   // q padded to 16 rows (rows 8-15 zero)
  __shared__ float attnL[16 * 512];  // unnormalized attn chunk
  __shared__ float Sloc[8];
  const int b = blockIdx.x, chunk = blockIdx.y;
  const int j0 = chunk * 512;
  const int t = threadIdx.x, lane = t & 31, w = t >> 5;
  const int mrow = lane & 15, klo = (lane >> 4) << 1;

  // Async stage q (8KB = 512 float4), zero pad rows with ds stores.
  {
    const float* gq = q + (long)b * 2048;
    for (int i = 0; i < 2; ++i) {
      int f4 = i * 256 + t;
      async_copy_b128(qpad + f4 * 4, gq + f4 * 4);
    }
  }
  for (int i = 0; i < 8; ++i) qpad[2048 + i * 256 + t] = 0.f;
  if (t < 8) Sloc[t] = 0.f;
  wait_asynccnt0();
  __syncthreads();

  float sacc[8];
  for (int r = 0; r < 8; ++r) sacc[r] = 0.f;

  // Phase 1: dots + slot-softmax; 4 j-tiles (16 j each) per wave
  for (int c = 0; c < 4; ++c) {
    int jt = w * 4 + c;
    long jb = (long)b * 8192 + j0 + jt * 16 + mrow;
    v8f acc = {};
    const float* qa = qpad + mrow * 256 + klo;
    const float* kp = kbuf + jb * 256 + klo;
    for (int kk = 0; kk < 256; kk += 4) {
      v2f af = *(const v2f*)(qa + kk);
      v2f bf = *(const v2f*)(kp + kk);
      acc = wmma_f32(af, bf, acc);
    }
    float e[8];
    if (lane < 16) {  // lanes 0-15 hold slots M=0..7 in regs 0..7
      float m = -1e30f;
      for (int r = 0; r < 8; ++r) {
        float dd = acc[r] * SCALEF;
        if (dd > m) m = dd;
      }
      float s = 0.f;
      for (int r = 0; r < 8; ++r) {
        e[r] = __expf(acc[r] * SCALEF - m);
        s += e[r];
      }
      float inv = 1.f / s;
      for (int r = 0; r < 8; ++r) {
        e[r] = e[r] * inv + EPSC;
        sacc[r] += e[r];
      }
    } else {  // padded slots M=8..15 -> zero
      for (int r = 0; r < 8; ++r) e[r] = 0.f;
    }
    int mbase = (lane >> 4) << 3;
    for (int r = 0; r < 8; ++r)
      attnL[(mbase + r) * 512 + jt * 16 + mrow] = e[r];
  }
  if (lane < 16)
    for (int r = 0; r < 8; ++r) atomicAdd(&Sloc[r], sacc[r]);
  __syncthreads();
  if (t < 8) atomicAdd(&Sbuf[b * 8 + t], Sloc[t]);

  // Phase 2: U_partial = attn(16x512) @ v(512x256); 2 col-tiles per wave
  for (int c = 0; c < 2; ++c) {
    int colbase = (w * 2 + c) * 16;
    v8f acc = {};
    const float* aa = attnL + mrow * 512 + klo;
    const float* vb = vbuf + ((long)b * 8192 + j0 + klo) * 256 + colbase + mrow;
    for (int kk = 0; kk < 512; kk += 4) {
      v2f af = *(const v2f*)(aa + kk);
      v2f bf;
      bf.x = vb[(long)kk * 256];        // B[k][n] = v[j0+k][col] (K-strided)
      bf.y = vb[(long)kk * 256 + 256];
      acc = wmma_f32(af, bf, acc);
    }
    if (lane < 16)  // valid slots M = r (0..7); padded rows are exact zeros
      for (int r = 0; r < 8; ++r)
        atomicAdd(&Ubuf[((long)b * 8 + r) * 256 + colbase + mrow], acc[r]);
  }
}

// ---------------------------------------------------------------------------
// Kernel 6: updates = U/S; GRU cell; residual MLP with LayerNorm. 1 block/row.
// ---------------------------------------------------------------------------
extern "C" __global__ void __launch_bounds__(256)
update_kernel(const float* __restrict__ U, const float* __restrict__ S,
              const float* __restrict__ W_ih, const float* __restrict__ W_hh,
              const float* __restrict__ b_ih, const float* __restrict__ b_hh,
              const float* __restrict__ W1, const float* __restrict__ b1,
              const float* __restrict__ W2, const float* __restrict__ b2,
              const float* __restrict__ g_ff, const float* __restrict__ be_ff,
              float* __restrict__ sets) {
  __shared__ float upd[256], hprev[256], hnew[256], xn[256], h1[256], red[256];
  int row = blockIdx.x, t = threadIdx.x;
  long base = (long)row * 256;
  upd[t] = U[base + t] / S[row];
  hprev[t] = sets[base + t];
  __syncthreads();
  float ar = b_ih[t], az = b_ih[256 + t], an = b_ih[512 + t];
  float hr = b_hh[t], hz = b_hh[256 + t], hn_ = b_hh[512 + t];
  const float* wi0 = W_ih + (long)t * 256;
  const float* wi1 = W_ih + (long)(256 + t) * 256;
  const float* wi2 = W_ih + (long)(512 + t) * 256;
  const float* wh0 = W_hh + (long)t * 256;
  const float* wh1 = W_hh + (long)(256 + t) * 256;
  const float* wh2 = W_hh + (long)(512 + t) * 256;
  for (int k = 0; k < 256; ++k) {
    float u = upd[k], hp = hprev[k];
    ar += u * wi0[k];
    az += u * wi1[k];
    an += u * wi2[k];
    hr += hp * wh0[k];
    hz += hp * wh1[k];
    hn_ += hp * wh2[k];
  }
  float r = 1.f / (1.f + __expf(-(ar + hr)));
  float z = 1.f / (1.f + __expf(-(az + hz)));
  float nn = tanhf(an + r * hn_);
  float h = (1.f - z) * nn + z * hprev[t];
  hnew[t] = h;
  __syncthreads();
  float mu = block_sum(h, red) * (1.f / 256.f);
  float d = h - mu;
  float var = block_sum(d * d, red) * (1.f / 256.f);
  xn[t] = d * rsqrtf(var + LNEPS) * g_ff[t] + be_ff[t];
  __syncthreads();
  float a1 = b1[t];
  const float* w1r = W1 + (long)t * 256;
  for (int k = 0; k < 256; ++k) a1 += xn[k] * w1r[k];
  h1[t] = a1 > 0.f ? a1 : 0.f;
  __syncthreads();
  float o = h + b2[t];
  const float* w2r = W2 + (long)t * 256;
  for (int k = 0; k < 256; ++k) o += h1[k] * w2r[k];
  sets[base + t] = o;
}

// ---------------------------------------------------------------------------
extern "C" void kernel_launch(void* const* d_in, const int* in_sizes, int n_in,
                              void* d_out, int out_size, void* d_ws,
                              size_t ws_size, hipStream_t stream) {
  const float* inputs = (const float*)d_in[0];
  const float* noise = (const float*)d_in[1];
  const float* sets_mu = (const float*)d_in[2];
  const float* sets_sigma = (const float*)d_in[3];
  const float* Wq = (const float*)d_in[4];
  const float* bq = (const float*)d_in[5];
  const float* Wk = (const float*)d_in[6];
  const float* bk = (const float*)d_in[7];
  const float* Wv = (const float*)d_in[8];
  const float* bv = (const float*)d_in[9];
  const float* W_ih = (const float*)d_in[10];
  const float* W_hh = (const float*)d_in[11];
  const float* b_ih = (const float*)d_in[12];
  const float* b_hh = (const float*)d_in[13];
  const float* W1 = (const float*)d_in[14];
  const float* b1 = (const float*)d_in[15];
  const float* W2 = (const float*)d_in[16];
  const float* b2 = (const float*)d_in[17];
  const float* g_in = (const float*)d_in[18];
  const float* be_in = (const float*)d_in[19];
  const float* g_sets = (const float*)d_in[20];
  const float* be_sets = (const float*)d_in[21];
  const float* g_ff = (const float*)d_in[22];
  const float* be_ff = (const float*)d_in[23];

  float* kbuf = (float*)d_ws;                          // 16*8192*256 f32
  float* vbuf = kbuf + (size_t)16 * 8192 * 256;        // 16*8192*256 f32
  float* qbuf = vbuf + (size_t)16 * 8192 * 256;        // 128*256
  float* Ubuf = qbuf + 128 * 256;                      // 128*256
  float* setsbuf = Ubuf + 128 * 256;                   // 128*256
  float* Sbuf = setsbuf + 128 * 256;                   // 128

  ln_kv_kernel<<<4096, 256, 0, stream>>>(inputs, g_in, be_in, Wk, bk, Wv, bv,
                                         kbuf, vbuf);
  init_sets_kernel<<<128, 256, 0, stream>>>(noise, sets_mu, sets_sigma,
                                            setsbuf);
  for (int it = 0; it < 5; ++it) {
    qproj_kernel<<<128, 256, 0, stream>>>(setsbuf, g_sets, be_sets, Wq, bq,
                                          qbuf);
    zero_us_kernel<<<128, 256, 0, stream>>>(Ubuf, Sbuf);
    attn_kernel<<<dim3(16, 16), 256, 0, stream>>>(qbuf, kbuf, vbuf, Ubuf,
                                                  Sbuf);
    update_kernel<<<128, 256, 0, stream>>>(Ubuf, Sbuf, W_ih, W_hh, b_ih, b_hh,
                                           W1, b1, W2, b2, g_ff, be_ff,
                                           setsbuf);
  }
  hipMemcpyAsync(d_out, setsbuf, (size_t)out_size * sizeof(float),
                 hipMemcpyDeviceToDevice, stream);
}